// SSIMLoss_30485677867343
// MI455X (gfx1250) — compile-verified
//
#include <hip/hip_runtime.h>

// SSIM loss, fused single pass over the images.
// 16x3x512x512 f32, window 11, separable box filter done as WMMA band-matrix
// matmuls (V_WMMA_F32_16X16X4_F32), SSIM math + deterministic reduction.

#define WIN      11
#define PAD      5
#define IMG_H    512
#define IMG_W    512
#define PLANES   48            // 16 * 3
#define TILE_R   16            // output rows per block
#define TILE_C   64            // output cols per block
#define HALO_R   28            // 16 + 10 halo rows, padded to K multiple of 4
#define HALO_C   74            // 64 + 10 halo cols
#define MID_R    16
#define MID_C    74
#define BLOCKS_PER_PLANE ((IMG_H / TILE_R) * (IMG_W / TILE_C))   // 32*8 = 256
#define NBLOCKS  (PLANES * BLOCKS_PER_PLANE)                      // 12288
#define NPIX     12582912.0f   // 16*3*512*512

typedef __attribute__((ext_vector_type(2))) float v2f;
typedef __attribute__((ext_vector_type(8))) float v8f;

__device__ __forceinline__ v8f wmma4(v2f a, v2f b, v8f c) {
  // 8-arg pattern per probe: (neg_a, A, neg_b, B, c_mod, C, reuse_a, reuse_b)
  return __builtin_amdgcn_wmma_f32_16x16x4_f32(false, a, false, b, (short)0, c,
                                               false, false);
}

__global__ __launch_bounds__(128) void ssim_tile_kernel(
    const float* __restrict__ img1, const float* __restrict__ img2,
    float* __restrict__ partial) {
  // 5 channels: 0:a 1:b 2:a*a 3:b*b 4:a*b
  __shared__ float s_prod[5][HALO_R][HALO_C];  // 41440 B
  __shared__ float s_mid[5][MID_R][MID_C];     // 23680 B
  __shared__ float s_wsum[4];

  const int tid  = threadIdx.x;
  const int wave = tid >> 5;       // wave32
  const int lane = tid & 31;
  const int n    = lane & 15;      // N-col for B/C/D, M-row for A
  const int hi   = lane >> 4;      // lane half selects K pair / D row half

  const int b      = blockIdx.x;
  const int plane  = b >> 8;                // / BLOCKS_PER_PLANE
  const int rem    = b & 255;
  const int row0   = (rem >> 3) * TILE_R;   // 32 row strips
  const int col0   = (rem & 7) * TILE_C;    // 8 col strips
  const float* p1 = img1 + (size_t)plane * (IMG_H * IMG_W);
  const float* p2 = img2 + (size_t)plane * (IMG_H * IMG_W);

  // ---------- Stage 1: halo load + pointwise products into LDS ----------
  for (int idx = tid; idx < HALO_R * HALO_C; idx += 128) {
    const int r  = idx / HALO_C;
    const int c  = idx - r * HALO_C;
    const int rg = row0 - PAD + r;
    const int cg = col0 - PAD + c;
    float va = 0.0f, vb = 0.0f;
    if (rg >= 0 && rg < IMG_H && cg >= 0 && cg < IMG_W) {
      const int off = rg * IMG_W + cg;
      va = p1[off];
      vb = p2[off];
    }
    s_prod[0][r][c] = va;
    s_prod[1][r][c] = vb;
    s_prod[2][r][c] = va * va;
    s_prod[3][r][c] = vb * vb;
    s_prod[4][r][c] = va * vb;
  }
  __syncthreads();

  const float inv = 1.0f / 11.0f;
  const v8f vzero = {0.f, 0.f, 0.f, 0.f, 0.f, 0.f, 0.f, 0.f};

  // ---------- Stage 2: vertical box filter = Band(16x28) x Prod(28x16) ----
  // 5 column tiles cover halo cols [0,74): c0 in {0,16,32,48,58} (last overlaps)
  for (int job = wave; job < 5; job += 4) {
    const int c0 = (job < 4) ? job * 16 : 58;
    v8f acc[5];
#pragma unroll
    for (int ch = 0; ch < 5; ++ch) acc[ch] = vzero;
#pragma unroll
    for (int t = 0; t < 7; ++t) {
      const int k0 = 4 * t + 2 * hi;  // this lane's K pair (A and B agree)
      // Band A[m][k] = inv iff k - m in [0,10]; m = lane&15
      v2f av;
      av[0] = ((unsigned)(k0 - n) <= 10u) ? inv : 0.0f;
      av[1] = ((unsigned)(k0 + 1 - n) <= 10u) ? inv : 0.0f;
#pragma unroll
      for (int ch = 0; ch < 5; ++ch) {
        v2f bv;
        bv[0] = s_prod[ch][k0][c0 + n];        // k0   <= 26 < HALO_R
        bv[1] = s_prod[ch][k0 + 1][c0 + n];    // k0+1 <= 27 < HALO_R
        acc[ch] = wmma4(av, bv, acc[ch]);
      }
    }
    // D layout: VGPR i -> row (hi*8 + i), col n
#pragma unroll
    for (int ch = 0; ch < 5; ++ch)
#pragma unroll
      for (int i = 0; i < 8; ++i) s_mid[ch][hi * 8 + i][c0 + n] = acc[ch][i];
  }
  __syncthreads();

  // ---------- Stage 3: horizontal box filter = Mid(16x28) x BandT(28x16) --
  {
    const int c0 = 16 * wave;  // output col tile, one per wave
    v8f acc[5];
#pragma unroll
    for (int ch = 0; ch < 5; ++ch) acc[ch] = vzero;
#pragma unroll
    for (int t = 0; t < 7; ++t) {
      const int k0 = 4 * t + 2 * hi;
      // Band B[k][nc] = inv iff k - nc in [0,10]
      v2f bv;
      bv[0] = ((unsigned)(k0 - n) <= 10u) ? inv : 0.0f;
      bv[1] = ((unsigned)(k0 + 1 - n) <= 10u) ? inv : 0.0f;
#pragma unroll
      for (int ch = 0; ch < 5; ++ch) {
        // A[m][k] = mid[m][c0+k]; clamp K-pad (k=26,27) to avoid OOB LDS;
        // band weight there is 0 anyway.
        v2f av;
        av[0] = (k0 < 26) ? s_mid[ch][n][c0 + k0] : 0.0f;
        av[1] = (k0 + 1 < 26) ? s_mid[ch][n][c0 + k0 + 1] : 0.0f;
        acc[ch] = wmma4(av, bv, acc[ch]);
      }
    }

    // ---------- Stage 4: SSIM map + reduction ----------
    const float C1 = 0.01f * 0.01f;
    const float C2 = 0.03f * 0.03f;
    float lsum = 0.0f;
#pragma unroll
    for (int i = 0; i < 8; ++i) {
      const float mu1 = acc[0][i], mu2 = acc[1][i];
      const float faa = acc[2][i], fbb = acc[3][i], fab = acc[4][i];
      const float mu1s = mu1 * mu1, mu2s = mu2 * mu2, mu12 = mu1 * mu2;
      const float s1 = faa - mu1s, s2 = fbb - mu2s, s12 = fab - mu12;
      const float num = (2.0f * mu12 + C1) * (2.0f * s12 + C2);
      const float den = (mu1s + mu2s + C1) * (s1 + s2 + C2);
      lsum += num / den;
    }
#pragma unroll
    for (int off = 16; off > 0; off >>= 1) lsum += __shfl_xor(lsum, off, 32);
    if (lane == 0) s_wsum[wave] = lsum;
  }
  __syncthreads();
  if (tid == 0) partial[b] = (s_wsum[0] + s_wsum[1]) + (s_wsum[2] + s_wsum[3]);
}

__global__ __launch_bounds__(256) void ssim_reduce_kernel(
    const float* __restrict__ partial, float* __restrict__ out) {
  __shared__ float s[256];
  float sum = 0.0f;
  for (int i = threadIdx.x; i < NBLOCKS; i += 256) sum += partial[i];
  s[threadIdx.x] = sum;
  __syncthreads();
  for (int off = 128; off > 0; off >>= 1) {
    if (threadIdx.x < off) s[threadIdx.x] += s[threadIdx.x + off];
    __syncthreads();
  }
  if (threadIdx.x == 0) out[0] = 1.0f - s[0] / NPIX;
}

extern "C" void kernel_launch(void* const* d_in, const int* in_sizes, int n_in,
                              void* d_out, int out_size, void* d_ws,
                              size_t ws_size, hipStream_t stream) {
  const float* img1 = (const float*)d_in[0];
  const float* img2 = (const float*)d_in[1];
  float* partial = (float*)d_ws;  // NBLOCKS floats = 48 KB, rewritten each call
  ssim_tile_kernel<<<NBLOCKS, 128, 0, stream>>>(img1, img2, partial);
  ssim_reduce_kernel<<<1, 256, 0, stream>>>(partial, (float*)d_out);
}